// DetectionLoss_82849919140443
// MI455X (gfx1250) — compile-verified
//
#include <hip/hip_runtime.h>
#include <stdint.h>

// ---------------------------------------------------------------- constants
namespace {
constexpr int B_ = 16, C_ = 3, A_ = 9, H_ = 128, W_ = 128, M_ = 32, K_ = 8;
constexpr int NPIX  = H_ * W_;       // 16384 pixels
constexpr int NANCH = A_ * NPIX;     // 147456 anchors per image

constexpr int BLK1  = 256;           // threads, kernel 1
constexpr int PIXPB = 256;           // pixels per block (1 per thread, 9 anchors each)
constexpr int NBLK1 = NPIX / PIXPB;  // 64 blocks per image
constexpr int BLK2  = 1024;          // threads, selection kernel

// workspace layout (32-bit words)
constexpr size_t WS_NEG = 0;                                   // B*NANCH u32
constexpr size_t WS_P1  = (size_t)B_ * NANCH;                  // B*NBLK1*5 f32
constexpr size_t WS_P2  = WS_P1 + (size_t)B_ * NBLK1 * 5;      // B*5 f32
} // namespace

// ---------------------------------------------------------------- CDNA5 async->LDS
// Toolchain declares these builtins with typed pointers:
//   b128: (global int __vector(4)*, lds int __vector(4)*, imm offset, imm cpol)
//   b32 : (global int*,             lds int*,             imm offset, imm cpol)
typedef int v4i_ __attribute__((vector_size(16)));
typedef __attribute__((address_space(1))) v4i_ g_v4i;
typedef __attribute__((address_space(3))) v4i_ l_v4i;
typedef __attribute__((address_space(1))) int  g_i32;
typedef __attribute__((address_space(3))) int  l_i32;

#if defined(__AMDGCN__) && __has_builtin(__builtin_amdgcn_global_load_async_to_lds_b128)
#define HAS_ASYNC128 1
#else
#define HAS_ASYNC128 0
#endif
#if defined(__AMDGCN__) && __has_builtin(__builtin_amdgcn_global_load_async_to_lds_b32)
#define HAS_ASYNC32 1
#else
#define HAS_ASYNC32 0
#endif

__device__ __forceinline__ void async_copy16(void* lds, const void* gmem) {
#if HAS_ASYNC128
  __builtin_amdgcn_global_load_async_to_lds_b128((g_v4i*)gmem, (l_v4i*)lds, 0, 0);
#else
  *reinterpret_cast<float4*>(lds) = *reinterpret_cast<const float4*>(gmem);
#endif
}
__device__ __forceinline__ void async_copy4(void* lds, const void* gmem) {
#if HAS_ASYNC32
  __builtin_amdgcn_global_load_async_to_lds_b32((g_i32*)gmem, (l_i32*)lds, 0, 0);
#else
  *reinterpret_cast<int*>(lds) = *reinterpret_cast<const int*>(gmem);
#endif
}
__device__ __forceinline__ void wait_async_zero() {
#if defined(__AMDGCN__)
#if __has_builtin(__builtin_amdgcn_s_wait_asynccnt)
  __builtin_amdgcn_s_wait_asynccnt(0);
#else
  asm volatile("s_wait_asynccnt 0" ::: "memory");
#endif
#endif
}

// ---------------------------------------------------------------- math helpers
__device__ __forceinline__ float bce_logits(float x, float t) {
  return fmaxf(x, 0.0f) - x * t + log1pf(expf(-fabsf(x)));
}
__device__ __forceinline__ float sl1(float d) {
  float ad = fabsf(d);
  return ad < 1.0f ? 0.5f * d * d : ad - 0.5f;
}

// ================================================================ kernel 1
// Per-anchor: IoU/argmax vs 32 GT boxes (LDS), obj BCE, sparse cls/loc losses.
// Writes: negbits[img*N + n] = bits(obj_loss) if neg else 0
//         part1[block*5]     = {pos_obj_sum, cls_sum, loc_sum, num_pos, num_neg}
__global__ __launch_bounds__(BLK1) void k_per_anchor(
    const float* __restrict__ pred, const float* __restrict__ anchors,
    const float* __restrict__ gtb,  const int*   __restrict__ gtl,
    uint32_t* __restrict__ negbits, float* __restrict__ part1)
{
  __shared__ float4 s_anc[PIXPB * A_];   // 2304 anchors, 36 KB
  __shared__ float4 s_gtb[M_];
  __shared__ int    s_gtl[M_];
  __shared__ float  s_red[5][BLK1];

  const int tid = (int)threadIdx.x;
  const int img = (int)blockIdx.x / NBLK1;
  const int blk = (int)blockIdx.x % NBLK1;
  const int pix = blk * PIXPB + tid;

  // ---- async-stage anchors + GT data into LDS (CDNA5 async copy) ----
  const float4* anc4 = reinterpret_cast<const float4*>(anchors);
  const int abase = blk * PIXPB * A_;
  #pragma unroll
  for (int i = 0; i < A_; ++i) {
    int idx = i * BLK1 + tid;                 // coalesced across lanes
    async_copy16(&s_anc[idx], &anc4[abase + idx]);
  }
  if (tid < M_) {
    async_copy16(&s_gtb[tid], &reinterpret_cast<const float4*>(gtb)[img * M_ + tid]);
    async_copy4(&s_gtl[tid], &gtl[img * M_ + tid]);
  }
  wait_async_zero();
  __syncthreads();

  const float*  pb = pred    + (size_t)img * (A_ * K_) * NPIX;
  uint32_t*     nb = negbits + (size_t)img * NANCH;

  float a0 = 0.f, a1 = 0.f, a2 = 0.f, a3 = 0.f, a4 = 0.f;

  for (int a = 0; a < A_; ++a) {
    const float4 an = s_anc[tid * A_ + a];
    const float areaA = (an.z - an.x) * (an.w - an.y);
    float best = -1.0f; int bidx = 0;
    #pragma unroll 8
    for (int j = 0; j < M_; ++j) {
      const float4 g = s_gtb[j];               // LDS broadcast
      float iw = fmaxf(fminf(an.z, g.z) - fmaxf(an.x, g.x), 0.f);
      float ih = fmaxf(fminf(an.w, g.w) - fmaxf(an.y, g.y), 0.f);
      float inter = iw * ih;
      float areaB = (g.z - g.x) * (g.w - g.y);
      float iou = inter / fmaxf(areaA + areaB - inter, 1e-9f);
      if (iou > best) { best = iou; bidx = j; }  // first-max like jnp.argmax
    }
    const bool pos = best >= 0.5f;
    const bool neg = best <  0.3f;
    const int ch = a * K_;

    const float pobj = pb[(ch + 4) * NPIX + pix];
    const float objl = bce_logits(pobj, pos ? 1.f : 0.f);

    a0 += pos ? objl : 0.f;
    a3 += pos ? 1.f : 0.f;
    a4 += neg ? 1.f : 0.f;
    nb[pix * A_ + a] = neg ? __float_as_uint(objl) : 0u;   // objl > 0 strictly

    if (pos) {   // sparse: only positives touch box/cls channels
      const float4 g = s_gtb[bidx];
      int tgt = s_gtl[bidx] - 1;
      tgt = tgt < 0 ? 0 : (tgt > C_ - 1 ? C_ - 1 : tgt);
      // classification: -log_softmax[tgt]
      float c0 = pb[(ch + 5) * NPIX + pix];
      float c1 = pb[(ch + 6) * NPIX + pix];
      float c2 = pb[(ch + 7) * NPIX + pix];
      float mx  = fmaxf(c0, fmaxf(c1, c2));
      float lse = mx + logf(expf(c0 - mx) + expf(c1 - mx) + expf(c2 - mx));
      float sel = (tgt == 0) ? c0 : ((tgt == 1) ? c1 : c2);
      a1 += lse - sel;
      // localization: smooth-L1 of (pred_box - encode(anchor, gt))
      float b0 = pb[(ch + 0) * NPIX + pix];
      float b1 = pb[(ch + 1) * NPIX + pix];
      float b2 = pb[(ch + 2) * NPIX + pix];
      float b3 = pb[(ch + 3) * NPIX + pix];
      float ax = (an.x + an.z) * 0.5f, ay = (an.y + an.w) * 0.5f;
      float aw = fmaxf(an.z - an.x, 1e-6f), ah = fmaxf(an.w - an.y, 1e-6f);
      float gx = (g.x + g.z) * 0.5f, gy = (g.y + g.w) * 0.5f;
      float gw = fmaxf(g.z - g.x, 1e-6f), gh = fmaxf(g.w - g.y, 1e-6f);
      a2 += sl1(b0 - (gx - ax) / aw) + sl1(b1 - (gy - ay) / ah)
          + sl1(b2 - logf(gw / aw)) + sl1(b3 - logf(gh / ah));
    }
  }

  // deterministic block reduction
  s_red[0][tid] = a0; s_red[1][tid] = a1; s_red[2][tid] = a2;
  s_red[3][tid] = a3; s_red[4][tid] = a4;
  __syncthreads();
  for (int s = BLK1 / 2; s > 0; s >>= 1) {
    if (tid < s) {
      #pragma unroll
      for (int m = 0; m < 5; ++m) s_red[m][tid] += s_red[m][tid + s];
    }
    __syncthreads();
  }
  if (tid == 0) {
    float* p = part1 + (size_t)blockIdx.x * 5;
    #pragma unroll
    for (int m = 0; m < 5; ++m) p[m] = s_red[m][0];
  }
}

// ================================================================ kernel 2
// Per image: exact top-k-sum of negative obj losses via 3-level radix select
// on float bit patterns (all losses > 0, so uint order == float order).
__global__ __launch_bounds__(BLK2) void k_select(
    const uint32_t* __restrict__ negbits, const float* __restrict__ part1,
    float* __restrict__ part2)
{
  __shared__ unsigned s_cnt[2048];
  __shared__ float    s_sum[BLK2];
  __shared__ float    s_meta[5];
  __shared__ int      s_k, s_rem;
  __shared__ unsigned s_pref;

  const int tid = (int)threadIdx.x;
  const int img = (int)blockIdx.x;
  const uint32_t* nb = negbits + (size_t)img * NANCH;

  if (tid == 0) {
    float s[5] = {0.f, 0.f, 0.f, 0.f, 0.f};
    const float* p = part1 + (size_t)img * NBLK1 * 5;
    for (int b = 0; b < NBLK1; ++b)
      for (int m = 0; m < 5; ++m) s[m] += p[b * 5 + m];
    for (int m = 0; m < 5; ++m) s_meta[m] = s[m];
    int np = (int)(s[3] + 0.5f), nn = (int)(s[4] + 0.5f);
    int k = (np == 0) ? ((nn > 0) ? ((nn / 10) > 1 ? (nn / 10) : 1) : 0)
                      : ((3 * np) < nn ? (3 * np) : nn);
    s_k = k; s_rem = k; s_pref = 0u;
  }
  __syncthreads();
  const int k = s_k;
  float topk = 0.f;

  if (k > 0) {
    for (int pass = 0; pass < 3; ++pass) {
      const int ls    = (pass == 0) ? 21 : (pass == 1 ? 10 : 0);
      const int nbins = (pass == 2) ? 1024 : 2048;
      for (int i = tid; i < 2048; i += BLK2) s_cnt[i] = 0u;
      __syncthreads();
      const unsigned pref = s_pref;
      for (int i = tid; i < NANCH; i += BLK2) {
        unsigned bits = nb[i];
        if (bits == 0u) continue;
        bool match = (pass == 0) ||
                     ((pass == 1 ? (bits >> 21) : (bits >> 10)) == pref);
        if (match) atomicAdd(&s_cnt[(bits >> ls) & (unsigned)(nbins - 1)], 1u);
      }
      __syncthreads();
      if (tid == 0) {
        int rem = s_rem; unsigned cum = 0; int selbin = 0;
        for (int b = nbins - 1; b >= 0; --b) {
          unsigned c = s_cnt[b];
          if (cum + c >= (unsigned)rem) { selbin = b; s_rem = rem - (int)cum; break; }
          cum += c;
        }
        s_pref = (pass == 2) ? ((s_pref << 10) | (unsigned)selbin)
                             : ((s_pref << 11) | (unsigned)selbin);
      }
      __syncthreads();
    }
    // deterministic sum of all values strictly greater than the pivot
    const unsigned pivot = s_pref;
    float local = 0.f;
    for (int i = tid; i < NANCH; i += BLK2) {
      unsigned bits = nb[i];
      local += (bits > pivot) ? __uint_as_float(bits) : 0.f;
    }
    s_sum[tid] = local;
    __syncthreads();
    for (int s = BLK2 / 2; s > 0; s >>= 1) {
      if (tid < s) s_sum[tid] += s_sum[tid + s];
      __syncthreads();
    }
    if (tid == 0) topk = s_sum[0] + (float)s_rem * __uint_as_float(pivot);
  }

  if (tid == 0) {
    float* o = part2 + (size_t)img * 5;
    o[0] = s_meta[0] + topk;          // obj_sum (pos + selected neg)
    o[1] = s_meta[1];                 // cls_sum
    o[2] = s_meta[2];                 // loc_sum
    o[3] = s_meta[3];                 // num_pos
    o[4] = s_meta[3] + (float)k;      // num_sel = num_pos + k
  }
}

// ================================================================ kernel 3
// Cross-image reduction on the matrix pipe: D += A_chunk(16x4) * ones(4x16),
// 4 chained V_WMMA_F32_16X16X4_F32 -> row m of D = total of metric m.
typedef __attribute__((ext_vector_type(2))) float v2f;
typedef __attribute__((ext_vector_type(8))) float v8f;

__global__ __launch_bounds__(32) void k_final(const float* __restrict__ p2,
                                              float* __restrict__ out)
{
  const int lane = (int)threadIdx.x;
  float t_obj, t_cls, t_loc, t_pos, t_sel;
#if defined(__AMDGCN__) && __has_builtin(__builtin_amdgcn_wmma_f32_16x16x4_f32)
  // A layout (16x4 f32): lanes 0-15 row M=lane, K = vgpr; lanes 16-31 K = 2+vgpr
  const int m    = lane & 15;
  const int mm   = (m < 5) ? m : 0;
  const float msk = (m < 5) ? 1.f : 0.f;     // branchless: EXEC stays all-ones
  const int kb   = (lane >> 4) << 1;
  v8f c = {0.f, 0.f, 0.f, 0.f, 0.f, 0.f, 0.f, 0.f};
  const v2f bones = {1.f, 1.f};              // ones matrix: layout-independent
  #pragma unroll
  for (int chunk = 0; chunk < 4; ++chunk) {
    const int i0 = chunk * 4 + kb;           // image index for K = kb
    v2f a;
    a.x = p2[i0 * 5 + mm] * msk;
    a.y = p2[(i0 + 1) * 5 + mm] * msk;
    c = __builtin_amdgcn_wmma_f32_16x16x4_f32(false, a, false, bones,
                                              (short)0, c, false, false);
  }
  // D row m, col 0 lives in lane 0, VGPR m (m = 0..4)
  t_obj = c[0]; t_cls = c[1]; t_loc = c[2]; t_pos = c[3]; t_sel = c[4];
#else
  t_obj = t_cls = t_loc = t_pos = t_sel = 0.f;
  for (int i = 0; i < B_; ++i) {
    t_obj += p2[i * 5 + 0]; t_cls += p2[i * 5 + 1]; t_loc += p2[i * 5 + 2];
    t_pos += p2[i * 5 + 3]; t_sel += p2[i * 5 + 4];
  }
#endif
  if (lane == 0) {
    const float denom_pos = fmaxf(t_pos, 1.0f);
    const float denom_obj = fmaxf(t_sel, 1.0f);
    const float loss_loc = t_loc / denom_pos;
    const float loss_cls = t_cls / denom_pos;
    const float loss_obj = t_obj / denom_obj;
    out[0] = loss_obj;
    out[1] = loss_cls;
    out[2] = loss_loc;
    out[3] = 2.0f * loss_loc + loss_cls + loss_obj;
  }
}

// ================================================================ launcher
extern "C" void kernel_launch(void* const* d_in, const int* in_sizes, int n_in,
                              void* d_out, int out_size, void* d_ws, size_t ws_size,
                              hipStream_t stream) {
  (void)in_sizes; (void)n_in; (void)out_size; (void)ws_size;
  const float* pred    = (const float*)d_in[0];
  const float* anchors = (const float*)d_in[1];
  const float* gtb     = (const float*)d_in[2];
  const int*   gtl     = (const int*)d_in[3];
  float* out = (float*)d_out;

  uint32_t* negbits = (uint32_t*)d_ws + WS_NEG;
  float*    part1   = (float*)d_ws + WS_P1;
  float*    part2   = (float*)d_ws + WS_P2;

  k_per_anchor<<<dim3(B_ * NBLK1), dim3(BLK1), 0, stream>>>(
      pred, anchors, gtb, gtl, negbits, part1);
  k_select<<<dim3(B_), dim3(BLK2), 0, stream>>>(negbits, part1, part2);
  k_final<<<dim3(1), dim3(32), 0, stream>>>(part2, out);
}